// NCEAverage_21208548508052
// MI455X (gfx1250) — compile-verified
//
#include <hip/hip_runtime.h>
#include <hip/hip_bf16.h>

// NCE Average (InstDisc memory bank) for gfx1250.
// probs[b,k] = exp(dot(bank[idx[b,k]], v[b]) / T) / Z,  Z = mean(probs)*NUM_IMAGES
// new_bank = bank with rows[image_index] = 0.5*old + 0.5*v
//
// Memory-bound problem (~1.85 GB traffic). Dot products computed with the
// full-precision f32 WMMA (V_WMMA_F32_16X16X4_F32): one wave = one batch b and
// 16 k-rows; 32 chained WMMAs cover H=128.

#define NUM_IMAGES 1281167
#define HIDDEN     128
#define KTOT       4097          // NUM_NOISE + 1
#define BATCH      256
#define INV_T      (1.0f / 0.07f)
#define TILES_PER_B 257          // ceil(4097/16)

typedef __attribute__((ext_vector_type(2))) float v2f;
typedef __attribute__((ext_vector_type(8))) float v8f;

// ---------------------------------------------------------------- zero Z acc
__global__ void zero_acc_kernel(float* acc) {
    if (threadIdx.x == 0 && blockIdx.x == 0) acc[0] = 0.0f;
}

// -------------------------------------------------- logits + exp + partial Z
// One wave handles (b, ktile): 16 k-rows. blockDim = 256 (8 waves/block).
__global__ void __launch_bounds__(256)
nce_logits_kernel(const float* __restrict__ image_vectors,
                  const float* __restrict__ memory_bank,
                  const int*   __restrict__ image_index,
                  const int*   __restrict__ sample_indices,
                  float*       __restrict__ probs_out,   // [BATCH*KTOT] unnormalized
                  float*       __restrict__ z_acc) {
    const int tid     = blockIdx.x * blockDim.x + threadIdx.x;
    const int wave    = tid >> 5;                 // global wave id == tile id
    const int lane    = threadIdx.x & 31;
    const int b       = wave / TILES_PER_B;
    const int kt      = wave % TILES_PER_B;
    const int k0      = kt * 16;

    const int row  = lane & 15;                   // M index 0..15
    const int half = lane >> 4;                   // 0: K=0,1  1: K=2,3

    // Resolve gathered row index for this lane's k-row (clamped for tail tile).
    int k  = k0 + row;
    int kk = (k < KTOT) ? k : (KTOT - 1);
    int idx = (kk == 0) ? image_index[b] : sample_indices[(long)b * KTOT + kk];

    const float* wrow = memory_bank   + (long)idx * HIDDEN;
    const float* vrow = image_vectors + (long)b   * HIDDEN;

    v8f acc = {};
#pragma unroll
    for (int h0 = 0; h0 < HIDDEN; h0 += 4) {
        // A[16x4] f32: lane (row,half) supplies A[row][2*half .. 2*half+1]
        v2f a = *(const v2f*)(wrow + h0 + 2 * half);
        // B[4x16] f32, every column = vector chunk: same per-lane addressing
        v2f bb = *(const v2f*)(vrow + h0 + 2 * half);
        // 8 args: (neg_a, A, neg_b, B, c_mod, C, reuse_a, reuse_b)
        acc = __builtin_amdgcn_wmma_f32_16x16x4_f32(
            false, a, false, bb, (short)0, acc, false, false);
    }

    // D layout: VGPR r = row r (lanes 0-15) / row r+8 (lanes 16-31).
    // All 16 columns identical -> lane 0 holds logits for rows 0..7,
    // lane 16 holds logits for rows 8..15.
    float partial = 0.0f;
    if ((lane & 15) == 0) {
#pragma unroll
        for (int r = 0; r < 8; ++r) {
            int m  = r + half * 8;
            int kw = k0 + m;
            if (kw < KTOT) {
                float p = __expf(acc[r] * INV_T);
                probs_out[(long)b * KTOT + kw] = p;   // unnormalized for now
                partial += p;
            }
        }
    }

    // Block reduction of partial sums -> one global atomic per block.
    __shared__ float sdata[256];
    sdata[threadIdx.x] = partial;
    __syncthreads();
#pragma unroll
    for (int s = 128; s > 0; s >>= 1) {
        if (threadIdx.x < s) sdata[threadIdx.x] += sdata[threadIdx.x + s];
        __syncthreads();
    }
    if (threadIdx.x == 0) atomicAdd(z_acc, sdata[0]);
}

// -------------------------------------------------------------- normalize
__global__ void __launch_bounds__(256)
normalize_kernel(float* __restrict__ probs, const float* __restrict__ z_acc) {
    const float sum = z_acc[0];
    // Z = (sum / (B*K)) * NUM_IMAGES ; probs /= Z
    const float scale = ((float)BATCH * (float)KTOT) /
                        (sum * (float)NUM_IMAGES);
    int i = blockIdx.x * blockDim.x + threadIdx.x;
    if (i < BATCH * KTOT) probs[i] *= scale;
}

// ------------------------------------------------------ bank streaming copy
__global__ void __launch_bounds__(256)
copy_bank_kernel(const float4* __restrict__ src, float4* __restrict__ dst,
                 long n4) {
    long i = (long)blockIdx.x * blockDim.x + threadIdx.x;
    if (i < n4) dst[i] = src[i];
}

// --------------------------------------------------------------- EMA scatter
__global__ void __launch_bounds__(128)
ema_update_kernel(const float* __restrict__ image_vectors,
                  const float* __restrict__ memory_bank,   // original (input)
                  const int*   __restrict__ image_index,
                  float*       __restrict__ bank_out) {
    const int b = blockIdx.x;        // 0..BATCH-1
    const int h = threadIdx.x;       // 0..HIDDEN-1
    const long row = (long)image_index[b] * HIDDEN + h;
    // MOMENTUM = 0.5 : new = 0.5*old + 0.5*v
    bank_out[row] = 0.5f * (memory_bank[row] + image_vectors[(long)b * HIDDEN + h]);
}

extern "C" void kernel_launch(void* const* d_in, const int* in_sizes, int n_in,
                              void* d_out, int out_size, void* d_ws, size_t ws_size,
                              hipStream_t stream) {
    const float* image_vectors  = (const float*)d_in[0];   // [256,128]
    const float* memory_bank    = (const float*)d_in[1];   // [1281167,128]
    const int*   image_index    = (const int*)d_in[2];     // [256]
    const int*   sample_indices = (const int*)d_in[3];     // [256,4097]

    float* probs    = (float*)d_out;                       // [256*4097]
    float* bank_out = (float*)d_out + (long)BATCH * KTOT;  // [1281167*128]
    float* z_acc    = (float*)d_ws;                        // 1 float scratch

    // 1) zero the Z accumulator (ws is not re-poisoned between replays)
    zero_acc_kernel<<<1, 1, 0, stream>>>(z_acc);

    // 2) WMMA logits + exp + partial Z sums
    const int total_waves = BATCH * TILES_PER_B;           // 65792
    const int blocks = total_waves / 8;                    // 8 waves per block
    nce_logits_kernel<<<blocks, 256, 0, stream>>>(
        image_vectors, memory_bank, image_index, sample_indices, probs, z_acc);

    // 3) normalize probs in place
    const int nprob = BATCH * KTOT;
    normalize_kernel<<<(nprob + 255) / 256, 256, 0, stream>>>(probs, z_acc);

    // 4) stream the memory bank into the output (float4 copy)
    const long n4 = (long)NUM_IMAGES * HIDDEN / 4;
    const long cblocks = (n4 + 255) / 256;
    copy_bank_kernel<<<(int)cblocks, 256, 0, stream>>>(
        (const float4*)memory_bank, (float4*)bank_out, n4);

    // 5) EMA scatter of the positive rows (after the copy, stream-ordered)
    ema_update_kernel<<<BATCH, HIDDEN, 0, stream>>>(
        image_vectors, memory_bank, image_index, bank_out);
}